// AdaptiveSignatureHedger_51960514347080
// MI455X (gfx1250) — compile-verified
//
#include <hip/hip_runtime.h>
#include <cmath>

#define B_    256
#define T_    1024
#define D_    5
#define W_    10
#define DC_   11
#define KSIG  132    // 11 + 121, multiple of 4
#define NSIG  256    // DEPTH * HID
#define KHEAD 72     // 69 padded to multiple of 4
#define HID_  64

typedef __attribute__((ext_vector_type(2))) float v2f;
typedef __attribute__((ext_vector_type(8))) float v8f;

static __device__ __forceinline__ int imax(int a, int b) { return a > b ? a : b; }

// ---------------------------------------------------------------------------
// Kernel 1: fold adaptive pooling into w_p -> effective (132 x 256) weight,
// and pad w_d1 (69x64 -> 72x64); both written in WMMA-B fragment order:
//   elem (k, n) stored at [((ks*2 + khalf)*N + n)*2 + e], k = ks*4 + khalf*2 + e
// ---------------------------------------------------------------------------
__global__ __launch_bounds__(256) void prep_kernel(const float* __restrict__ w_p,
                                                   const float* __restrict__ w_d1,
                                                   float* __restrict__ weffB,
                                                   float* __restrict__ wd1B) {
    int blk = blockIdx.x;
    if (blk < KSIG) {
        int j = blk;                 // sig2 channel 0..131
        int col = threadIdx.x;       // d*64 + h
        int d = col >> 6, h = col & 63;
        int n = (d == 0) ? DC_ : KSIG;   // d==0 pools level1 (11 ch), else sig2 (132 ch)
        float acc = 0.0f;
        if (!(d == 0 && j >= DC_)) {
            for (int i = 0; i < 256; ++i) {
                int s = (i * n) >> 8;
                int e = ((i + 1) * n + 255) >> 8;
                float wv = w_p[((size_t)d * 256 + i) * 64 + h];
                if (e - s == 2) {
                    if (j == s || j == s + 1) acc += 0.5f * wv;
                } else {
                    if (j == s) acc += wv;
                }
            }
        }
        int ks = j >> 2, rem = j & 3, hf = rem >> 1, e2 = rem & 1;
        weffB[(((ks * 2 + hf) * NSIG) + col) * 2 + e2] = acc;
    } else {
        int k = blk - KSIG;          // 0..71 padded head row
        int col = threadIdx.x;
        if (col < HID_) {
            float v = (k < HID_ + D_) ? w_d1[k * HID_ + col] : 0.0f;
            int ks = k >> 2, rem = k & 3, hf = rem >> 1, e2 = rem & 1;
            wd1B[(((ks * 2 + hf) * HID_) + col) * 2 + e2] = v;
        }
    }
}

// ---------------------------------------------------------------------------
// Kernel 2: per batch row: returns, scanned |returns| -> vol, h-MLP, regime
// MLP + softmax -> gate weights w (B,T,4)
// ---------------------------------------------------------------------------
__global__ __launch_bounds__(128) void gates_kernel(const float* __restrict__ feat,
                                                    const float* __restrict__ w_h1,
                                                    const float* __restrict__ b_h1,
                                                    const float* __restrict__ w_h2,
                                                    const float* __restrict__ b_h2,
                                                    const float* __restrict__ w_g1,
                                                    const float* __restrict__ b_g1,
                                                    const float* __restrict__ w_g2,
                                                    const float* __restrict__ b_g2,
                                                    float* __restrict__ gates) {
    __shared__ float s_ret[T_];
    __shared__ float s_cs[T_];
    __shared__ float s_part[128];
    __shared__ float s_wh1[W_ * 32];
    __shared__ float s_bh1[32], s_wh2[32], s_wg1[64], s_bg1[32], s_wg2[128], s_bg2[4];

    int tid = threadIdx.x;
    int b = blockIdx.x;

    for (int i = tid; i < W_ * 32; i += 128) s_wh1[i] = w_h1[i];
    if (tid < 32) { s_bh1[tid] = b_h1[tid]; s_wh2[tid] = w_h2[tid]; s_bg1[tid] = b_g1[tid]; }
    if (tid < 64) s_wg1[tid] = w_g1[tid];
    s_wg2[tid] = w_g2[tid];            // 128 entries, one per thread
    if (tid < 4) s_bg2[tid] = b_g2[tid];

    for (int t = tid; t < T_; t += 128) {
        float f = feat[((size_t)b * T_ + t) * D_];
        float fp = (t > 0) ? feat[((size_t)b * T_ + t - 1) * D_] : f;
        s_ret[t] = f - fp;
    }
    __syncthreads();

    // inclusive scan of |ret|: 8 contiguous elems per thread + Hillis-Steele
    float loc[8];
    float run = 0.0f;
    int base = tid * 8;
#pragma unroll
    for (int i = 0; i < 8; ++i) { run += fabsf(s_ret[base + i]); loc[i] = run; }
    s_part[tid] = run;
    __syncthreads();
    for (int off = 1; off < 128; off <<= 1) {
        float v = s_part[tid];
        float add = (tid >= off) ? s_part[tid - off] : 0.0f;
        __syncthreads();
        s_part[tid] = v + add;
        __syncthreads();
    }
    float prev = (tid > 0) ? s_part[tid - 1] : 0.0f;
#pragma unroll
    for (int i = 0; i < 8; ++i) s_cs[base + i] = prev + loc[i];
    __syncthreads();

    float bh2 = b_h2[0];
    for (int t = tid; t < T_; t += 128) {
        int tp = imax(t - W_, 0);
        float rw[W_];
#pragma unroll
        for (int k = 0; k < W_; ++k) rw[k] = s_ret[tp + k];
        // h MLP: 10 -> 32 -> 1
        float u = bh2;
        for (int j = 0; j < 32; ++j) {
            float z = s_bh1[j];
#pragma unroll
            for (int k = 0; k < W_; ++k) z += rw[k] * s_wh1[k * 32 + j];
            u += fmaxf(z, 0.0f) * s_wh2[j];
        }
        float H = 0.5f / (1.0f + expf(-u));
        float vol = s_cs[t] / ((float)(t + 1) + 1e-8f);
        // regime MLP: 2 -> 32 -> 4, softmax
        float lg[4];
#pragma unroll
        for (int d = 0; d < 4; ++d) lg[d] = s_bg2[d];
        for (int j = 0; j < 32; ++j) {
            float z = fmaxf(H * s_wg1[j] + vol * s_wg1[32 + j] + s_bg1[j], 0.0f);
#pragma unroll
            for (int d = 0; d < 4; ++d) lg[d] += z * s_wg2[j * 4 + d];
        }
        float m = fmaxf(fmaxf(lg[0], lg[1]), fmaxf(lg[2], lg[3]));
        float es[4], ssum = 0.0f;
#pragma unroll
        for (int d = 0; d < 4; ++d) { es[d] = expf(lg[d] - m); ssum += es[d]; }
        float inv = 1.0f / ssum;
        float4 g4 = make_float4(es[0] * inv, es[1] * inv, es[2] * inv, es[3] * inv);
        reinterpret_cast<float4*>(gates)[(size_t)b * T_ + t] = g4;
    }
}

// ---------------------------------------------------------------------------
// Kernel 3: signature construction + WMMA GEMMs + head.
// 128 threads = 4 waves; each wave handles 16 consecutive t of one batch row.
// ---------------------------------------------------------------------------
__global__ __launch_bounds__(128) void main_kernel(const float* __restrict__ feat,
                                                   const float* __restrict__ gates,
                                                   const float* __restrict__ weffB,
                                                   const float* __restrict__ wd1B,
                                                   const float* __restrict__ b_p,
                                                   const float* __restrict__ b_d1,
                                                   const float* __restrict__ w_d2,
                                                   const float* __restrict__ b_d2,
                                                   float* __restrict__ out) {
    __shared__ float s_inc[4][16][9][12];   // increments, cross-lane for outer prod
    __shared__ float s_A[4][16][KSIG];      // sig2 rows (A matrix); reused for head A
    __shared__ float s_fw[4][26][D_];       // staged feature window

    int wv = threadIdx.x >> 5;
    int lane = threadIdx.x & 31;
    int b = blockIdx.x >> 4;                // T/64 = 16 blocks of t per batch row
    int t0 = (blockIdx.x & 15) * 64 + wv * 16;

    // ---- stage feature window [max(t0-10,0), t0+15] (contiguous in memory) ----
    int ttmin = imax(t0 - W_, 0);
    int nwin = (t0 + 15) - ttmin + 1;       // 16 or 26
    const float* fbase = feat + ((size_t)b * T_ + ttmin) * D_;
    for (int idx = lane; idx < nwin * D_; idx += 32)
        (&s_fw[wv][0][0])[idx] = fbase[idx];
    __syncthreads();

    // ---- phase 1: per-row normalized increments (2 lanes per row) ----
    int r = lane >> 1, hl = lane & 1;
    int t = t0 + r;
    int Lp = (t < W_) ? t : W_;
    float invden = 1.0f / (float)imax(Lp - 1, 1);
    int basew = imax(t - W_, 0);
    int nch = hl ? 5 : 6;
    int ch0 = hl ? 6 : 0;

    float increg[6][9];
#pragma unroll
    for (int ci = 0; ci < 6; ++ci) {
        if (ci < nch) {
            int ch = ch0 + ci;
            float x[10];
            if (ch < 2 * D_) {
                int cc = (ch < D_) ? ch : (ch - D_);
                int kadd = (ch < D_) ? 0 : 1;
#pragma unroll
                for (int k = 0; k < W_; ++k) {
                    int tt = basew + k + kadd; if (tt > t) tt = t;
                    x[k] = s_fw[wv][tt - ttmin][cc];
                }
            } else {
#pragma unroll
                for (int k = 0; k < W_; ++k) x[k] = (float)k * invden;
            }
            float sum = 0.0f;
#pragma unroll
            for (int k = 0; k < W_; ++k) sum += (k < Lp) ? x[k] : 0.0f;
            float mean = sum / (float)imax(Lp, 1);
            float vs = 0.0f;
#pragma unroll
            for (int k = 0; k < W_; ++k) { float dx = x[k] - mean; vs += (k < Lp) ? dx * dx : 0.0f; }
            float inv = 1.0f / (sqrtf(vs * invden) + 1e-6f);   // mean cancels in diffs
            float lvl = 0.0f;
#pragma unroll
            for (int k = 0; k < W_ - 1; ++k) {
                float iv = (k < Lp - 1) ? (x[k + 1] - x[k]) * inv : 0.0f;
                increg[ci][k] = iv;
                s_inc[wv][r][k][ch] = iv;
                lvl += iv;
            }
            s_A[wv][r][ch] = lvl;           // level1 channels = sig2[0..10]
        } else {
#pragma unroll
            for (int k = 0; k < W_ - 1; ++k) increg[ci][k] = 0.0f;
        }
    }
    __syncthreads();

    // ---- outer products -> sig2[11 + i*11 + j] ----
    for (int j = 0; j < DC_; ++j) {
        float xj[9];
#pragma unroll
        for (int k = 0; k < 9; ++k) xj[k] = s_inc[wv][r][k][j];
#pragma unroll
        for (int ci = 0; ci < 6; ++ci) {
            if (ci < nch) {
                int i = ch0 + ci;
                float a = 0.0f;
#pragma unroll
                for (int k = 0; k < 9; ++k) a += increg[ci][k] * xj[k];
                s_A[wv][r][DC_ + i * DC_ + j] = a;
            }
        }
    }
    __syncthreads();

    // ---- main GEMM: (16 x 132) @ (132 x 256), f32 WMMA 16x16x4 ----
    int li = lane & 15;                      // M row for A, N col within tile for B/D
    int hv = lane >> 4;                      // K-half selector (A/B), M-half for D
    v8f zero = {0.f, 0.f, 0.f, 0.f, 0.f, 0.f, 0.f, 0.f};
    v8f acc[16];
#pragma unroll
    for (int nt = 0; nt < 16; ++nt) acc[nt] = zero;

    const float* Arow = &s_A[wv][li][0];
    for (int ks = 0; ks < KSIG / 4; ++ks) {
        int kb = ks * 4 + hv * 2;
        v2f a; a.x = Arow[kb]; a.y = Arow[kb + 1];
        const float* Bbase = weffB + ((size_t)(ks * 2 + hv) * NSIG) * 2;
#pragma unroll
        for (int nt = 0; nt < 16; ++nt) {
            v2f bb = *(const v2f*)(Bbase + (nt * 16 + li) * 2);
            acc[nt] = __builtin_amdgcn_wmma_f32_16x16x4_f32(false, a, false, bb,
                                                            (short)0, acc[nt], false, false);
        }
    }

    // ---- gates, bias, relu, depth-weighted sum -> sig_repr; restage head A ----
    float gate[8][4];
#pragma unroll
    for (int e = 0; e < 8; ++e) {
        int M = e + hv * 8;
        float4 g = reinterpret_cast<const float4*>(gates)[(size_t)b * T_ + t0 + M];
        gate[e][0] = g.x; gate[e][1] = g.y; gate[e][2] = g.z; gate[e][3] = g.w;
    }
    __syncthreads();
#pragma unroll
    for (int c = 0; c < 4; ++c) {
        int hcol = c * 16 + li;
        float bp[4];
#pragma unroll
        for (int d = 0; d < 4; ++d) bp[d] = b_p[d * HID_ + hcol];
#pragma unroll
        for (int e = 0; e < 8; ++e) {
            float srep = 0.0f;
#pragma unroll
            for (int d = 0; d < 4; ++d)
                srep += gate[e][d] * fmaxf(acc[d * 4 + c][e] + bp[d], 0.0f);
            s_A[wv][e + hv * 8][hcol] = srep;           // head A cols 0..63
        }
    }
    // head A cols 64..71: features (5) + zero pad (3)
#pragma unroll
    for (int ii = 0; ii < 4; ++ii) {
        int flat = lane + 32 * ii;                       // 0..127
        int rr = flat >> 3, cc = flat & 7;
        float val = (cc < D_) ? feat[((size_t)b * T_ + t0 + rr) * D_ + cc] : 0.0f;
        s_A[wv][rr][HID_ + cc] = val;
    }
    __syncthreads();

    // ---- head GEMM: (16 x 72) @ (72 x 64) ----
    v8f acc2[4];
#pragma unroll
    for (int nt = 0; nt < 4; ++nt) acc2[nt] = zero;
    for (int ks = 0; ks < KHEAD / 4; ++ks) {
        int kb = ks * 4 + hv * 2;
        v2f a; a.x = Arow[kb]; a.y = Arow[kb + 1];
        const float* Bbase = wd1B + ((size_t)(ks * 2 + hv) * HID_) * 2;
#pragma unroll
        for (int nt = 0; nt < 4; ++nt) {
            v2f bb = *(const v2f*)(Bbase + (nt * 16 + li) * 2);
            acc2[nt] = __builtin_amdgcn_wmma_f32_16x16x4_f32(false, a, false, bb,
                                                             (short)0, acc2[nt], false, false);
        }
    }

    // ---- relu + dot with w_d2, cross-lane reduce, tanh ----
    float p[8];
#pragma unroll
    for (int e = 0; e < 8; ++e) p[e] = 0.0f;
#pragma unroll
    for (int nt = 0; nt < 4; ++nt) {
        int hcol = nt * 16 + li;
        float bd = b_d1[hcol], wd = w_d2[hcol];
#pragma unroll
        for (int e = 0; e < 8; ++e)
            p[e] += fmaxf(acc2[nt][e] + bd, 0.0f) * wd;
    }
    float* part = &s_inc[wv][0][0][0];                   // reuse: 16 rows x 16 lanes
#pragma unroll
    for (int e = 0; e < 8; ++e) part[(e + hv * 8) * 16 + li] = p[e];
    __syncthreads();
    if (lane < 16) {
        float s = b_d2[0];
#pragma unroll
        for (int q = 0; q < 16; ++q) s += part[lane * 16 + q];
        out[(size_t)b * T_ + t0 + lane] = 1.5f * tanhf(s);
    }
}

// ---------------------------------------------------------------------------
extern "C" void kernel_launch(void* const* d_in, const int* in_sizes, int n_in,
                              void* d_out, int out_size, void* d_ws, size_t ws_size,
                              hipStream_t stream) {
    const float* feat = (const float*)d_in[0];
    const float* w_h1 = (const float*)d_in[1];
    const float* b_h1 = (const float*)d_in[2];
    const float* w_h2 = (const float*)d_in[3];
    const float* b_h2 = (const float*)d_in[4];
    const float* w_g1 = (const float*)d_in[5];
    const float* b_g1 = (const float*)d_in[6];
    const float* w_g2 = (const float*)d_in[7];
    const float* b_g2 = (const float*)d_in[8];
    const float* w_p  = (const float*)d_in[9];
    const float* b_p  = (const float*)d_in[10];
    const float* w_d1 = (const float*)d_in[11];
    const float* b_d1 = (const float*)d_in[12];
    const float* w_d2 = (const float*)d_in[13];
    const float* b_d2 = (const float*)d_in[14];
    float* outp = (float*)d_out;

    float* ws    = (float*)d_ws;
    float* gatesW = ws;                                   // B*T*4   = 1048576 floats
    float* weffB  = ws + (size_t)B_ * T_ * 4;             // 33*2*256*2 = 33792 floats
    float* wd1B   = weffB + 33 * 2 * NSIG * 2;            // 18*2*64*2  = 4608 floats

    prep_kernel<<<KSIG + KHEAD, 256, 0, stream>>>(w_p, w_d1, weffB, wd1B);
    gates_kernel<<<B_, 128, 0, stream>>>(feat, w_h1, b_h1, w_h2, b_h2,
                                         w_g1, b_g1, w_g2, b_g2, gatesW);
    main_kernel<<<(B_ * T_) / 64, 128, 0, stream>>>(feat, gatesW, weffB, wd1B,
                                                    b_p, b_d1, w_d2, b_d2, outp);
}